// Parser_20409684591404
// MI455X (gfx1250) — compile-verified
//
#include <hip/hip_runtime.h>
#include <math.h>
#include <stdint.h>

#define NMOL     512
#define MOLSIZE  128
#define CUT2     (144.0f)          // OUTERCUTOFF^2
#define LCF_F    (1.8897259886f)

typedef __attribute__((ext_vector_type(2))) float v2f;
typedef __attribute__((ext_vector_type(8))) float v8f;

// Stage one molecule's coordinates (MOLSIZE x 3 f32, 12 B/atom in global) into
// LDS laid out as [MOLSIZE][4] f32 (16 B/atom) using CDNA5 async global->LDS
// copies (ASYNCcnt path). INST_OFFSET is added to BOTH the LDS and the global
// address (CDNA5 ISA), so per-component spacing (4 B on both
// sides) rides the offset: modifier while the per-atom strides (16 B LDS vs
// 12 B global) live in the two VGPR offsets.
__device__ __forceinline__ void async_stage_coords(float (*X)[4], int t,
                                                   const float* gbase)
{
    const unsigned ldsOff = (unsigned)(uintptr_t)(&X[t][0]); // low 32b = LDS byte addr
    const unsigned vOff   = (unsigned)(t * 12);
    asm volatile(
        "global_load_async_to_lds_b32 %0, %1, %2\n\t"
        "global_load_async_to_lds_b32 %0, %1, %2 offset:4\n\t"
        "global_load_async_to_lds_b32 %0, %1, %2 offset:8"
        :: "v"(ldsOff), "v"(vOff), "s"(gbase) : "memory");
}
__device__ __forceinline__ void async_wait_all()
{
    asm volatile("s_wait_asynccnt 0x0" ::: "memory");
}

// ---------------------------------------------------------------------------
// K1: per-molecule. Async coord staging -> WMMA f32 16x16x4 Gram tiles ->
// pair predicate bitmask in LDS (ballot + ds_or), pair/atom counts, stats.
// ---------------------------------------------------------------------------
__global__ __launch_bounds__(MOLSIZE)
void parser_k1(const int* __restrict__ species, const float* __restrict__ coords,
               const int* __restrict__ totq, const float* __restrict__ tore,
               float* __restrict__ out, unsigned* __restrict__ bm,
               int* __restrict__ pairCnt, int* __restrict__ atomCnt)
{
    __shared__ float    X[MOLSIZE][4];     // xyz0 (K-padded for WMMA)
    __shared__ float    n2[MOLSIZE];       // |c|^2
    __shared__ int      nbS[MOLSIZE];      // nonblank flag
    __shared__ unsigned msk[MOLSIZE][4];   // 128-bit predicate per row
    __shared__ int      ri[MOLSIZE];       // reduce scratch (int)
    __shared__ float    rf[MOLSIZE];       // reduce scratch (float)

    const int m = blockIdx.x;
    const int t = threadIdx.x;

    // async stage coords while we do the scalar-ish work
    async_stage_coords(X, t, coords + (size_t)m * MOLSIZE * 3);

    const int s = species[m * MOLSIZE + t];
    __builtin_prefetch(&tore[s], 0, 0);                 // global_prefetch_b8
    X[t][3] = 0.0f;                                     // K=3 pad (not async-written)
    nbS[t]  = (s > 0);
    msk[t][0] = 0u; msk[t][1] = 0u; msk[t][2] = 0u; msk[t][3] = 0u;

    async_wait_all();
    __syncthreads();

    const float x = X[t][0], y = X[t][1], z = X[t][2];
    n2[t] = x * x + y * y + z * z;
    __syncthreads();

    // ---- WMMA tiles: G[i][j] = ci . cj for upper-triangular 16x16 tiles ----
    const int wave = t >> 5;
    const int lane = t & 31;
    const int mrow = lane & 15;
    const int kb   = (lane >> 4) << 1;   // K pair base: 0 or 2
    const int hi8  = (lane >> 4) << 3;   // C/D row offset: 0 or 8

    for (int tile = wave; tile < 64; tile += 4) {
        const int ti = tile >> 3;
        const int tj = tile & 7;
        if (ti > tj) continue;           // wave-uniform at runtime; EXEC full for WMMA

        v2f a, b;
        const int ia = ti * 16 + mrow;   // A: rows = atoms i
        const int jb = tj * 16 + mrow;   // B: cols = atoms j (mirrored gather)
        a.x = X[ia][kb]; a.y = X[ia][kb + 1];
        b.x = X[jb][kb]; b.y = X[jb][kb + 1];
        v8f c = {};
        c = __builtin_amdgcn_wmma_f32_16x16x4_f32(
                /*neg_a=*/false, a, /*neg_b=*/false, b,
                /*c_mod=*/(short)0, c, /*reuse_a=*/false, /*reuse_b=*/false);

        const int   col = tj * 16 + (lane & 15);
        const float njn = n2[col];
        const int   nbj = nbS[col];
        const int   wsel = tj >> 1;
        const int   sh   = (tj & 1) << 4;

        #pragma unroll
        for (int v = 0; v < 8; ++v) {
            const int   row = ti * 16 + v + hi8;
            const float d2  = n2[row] + njn - 2.0f * c[v];
            // bitwise & keeps this branch-free (no exec save/restore per v)
            const int pred = (int)(col > row) & nbS[row] & nbj & (int)(d2 < CUT2);
            const unsigned bal = __builtin_amdgcn_ballot_w32(pred != 0);
            if (lane == 0) {
                atomicOr(&msk[ti * 16 + v    ][wsel], (bal & 0xFFFFu) << sh);
                atomicOr(&msk[ti * 16 + v + 8][wsel], (bal >> 16)     << sh);
            }
        }
    }
    __syncthreads();

    // ---- spill bitmask + per-row counts ----
    const unsigned w0 = msk[t][0], w1 = msk[t][1], w2 = msk[t][2], w3 = msk[t][3];
    uint4* gbm = (uint4*)&bm[(m * MOLSIZE + t) * 4];
    *gbm = make_uint4(w0, w1, w2, w3);
    const int cnt = __popc(w0) + __popc(w1) + __popc(w2) + __popc(w3);

    // ---- block reductions ----
#define REDUCE_I(val, res) do { __syncthreads(); ri[t] = (val); __syncthreads();     \
        for (int o = 64; o > 0; o >>= 1) { if (t < o) ri[t] += ri[t + o];            \
            __syncthreads(); } (res) = ri[0]; } while (0)

    const int nb    = (s > 0);
    const int heavy = (s > 1);
    const int hydro = (s == 1);
    const int sh2   = ((s > 12) && (s < 18)) || ((s > 20) && (s < 30)) ||
                      ((s > 32) && (s < 36)) || ((s > 38) && (s < 48)) ||
                      ((s > 50) && (s < 54)) || ((s > 70) && (s < 80)) || (s == 57);
    const float chg = tore[s];

    int pairTot, atomTot, shTot, heavyTot, hydroTot;
    REDUCE_I(cnt,   pairTot);
    REDUCE_I(nb,    atomTot);
    REDUCE_I(sh2,   shTot);
    REDUCE_I(heavy, heavyTot);
    REDUCE_I(hydro, hydroTot);
#undef REDUCE_I

    __syncthreads(); rf[t] = chg; __syncthreads();
    for (int o = 64; o > 0; o >>= 1) { if (t < o) rf[t] += rf[t + o]; __syncthreads(); }
    const float chgTot = rf[0];

    if (t == 0) {
        pairCnt[m] = pairTot;
        atomCnt[m] = atomTot;
        out[2 + m]            = (float)shTot;     // nSuperHeavy
        out[2 + NMOL + m]     = (float)heavyTot;  // nHeavy
        out[2 + 2*NMOL + m]   = (float)hydroTot;  // nHydro
        const int nchg = (int)chgTot - totq[m];   // astype(int32) truncates toward 0
        const int nocc = (nchg >= 0) ? (nchg >> 1) : -((-nchg + 1) >> 1); // floor div
        out[2 + 3*NMOL + m]   = (float)nocc;      // nocc
    }
}

// ---------------------------------------------------------------------------
// K2: single block. Exclusive scans over 512 molecule counts -> bases/totals.
// ---------------------------------------------------------------------------
__global__ __launch_bounds__(NMOL)
void parser_k2(float* __restrict__ out,
               const int* __restrict__ pairCnt, const int* __restrict__ atomCnt,
               int* __restrict__ pairBase, int* __restrict__ atomBase,
               int* __restrict__ totals)
{
    __shared__ int sp[NMOL], sa[NMOL];
    const int t = threadIdx.x;
    const int p0 = pairCnt[t], a0 = atomCnt[t];
    sp[t] = p0; sa[t] = a0;
    __syncthreads();
    for (int o = 1; o < NMOL; o <<= 1) {
        int vp = 0, va = 0;
        if (t >= o) { vp = sp[t - o]; va = sa[t - o]; }
        __syncthreads();
        sp[t] += vp; sa[t] += va;
        __syncthreads();
    }
    pairBase[t] = sp[t] - p0;
    atomBase[t] = sa[t] - a0;
    if (t == NMOL - 1) { totals[0] = sa[t]; totals[1] = sp[t]; }  // n_real, npairs
    if (t == 0)        { out[0] = (float)NMOL; out[1] = (float)MOLSIZE; }
}

// ---------------------------------------------------------------------------
// K3: per-molecule. Atom compaction (Z, maskd, atom_molid) and ordered pair
// record emission from the stored bitmask, at device-computed offsets.
// ---------------------------------------------------------------------------
__global__ __launch_bounds__(MOLSIZE)
void parser_k3(const int* __restrict__ species, const float* __restrict__ coords,
               float* __restrict__ out, const unsigned* __restrict__ bm,
               const int* __restrict__ pairBase, const int* __restrict__ atomBase,
               const int* __restrict__ totals)
{
    __shared__ float X[MOLSIZE][4];
    __shared__ int   spec[MOLSIZE];
    __shared__ int   cIdx[MOLSIZE];   // compact (global) atom index
    __shared__ int   scan[MOLSIZE];

    const int m = blockIdx.x;
    const int t = threadIdx.x;

    async_stage_coords(X, t, coords + (size_t)m * MOLSIZE * 3);

    const int R = totals[0];          // n_real
    const int P = totals[1];          // npairs
    const int offZ     = 2 + 4 * NMOL;
    const int offMaskd = offZ     + R;
    const int offAmol  = offMaskd + R;
    const int offMask  = offAmol  + R;
    const int offMaskL = offMask  + P;
    const int offPmol  = offMaskL + P;
    const int offNi    = offPmol  + P;
    const int offNj    = offNi    + P;
    const int offIdxi  = offNj    + P;
    const int offIdxj  = offIdxi  + P;
    const int offXij   = offIdxj  + P;
    const int offRij   = offXij   + 3 * P;

    const int s = species[m * MOLSIZE + t];
    spec[t] = s;

    // ---- atom compaction: exclusive scan of nonblank flags ----
    const int nb = (s > 0);
    scan[t] = nb;
    async_wait_all();                 // coords resident before first barrier
    __syncthreads();
    for (int o = 1; o < MOLSIZE; o <<= 1) {
        int v = 0; if (t >= o) v = scan[t - o];
        __syncthreads();
        scan[t] += v;
        __syncthreads();
    }
    const int rank = scan[t] - nb;
    const int ci   = atomBase[m] + rank;
    cIdx[t] = ci;
    if (nb) {
        out[offZ + ci]     = (float)s;
        out[offMaskd + ci] = (float)(t * (MOLSIZE + 1) + m * MOLSIZE * MOLSIZE);
        out[offAmol + ci]  = (float)m;
    }

    const float xi = X[t][0], yi = X[t][1], zi = X[t][2];

    // ---- per-row pair counts -> exclusive scan ----
    const uint4 w = *(const uint4*)&bm[(m * MOLSIZE + t) * 4];
    unsigned words[4] = { w.x, w.y, w.z, w.w };
    const int cnt = __popc(w.x) + __popc(w.y) + __popc(w.z) + __popc(w.w);
    __syncthreads();                 // protect scan[] reuse (also fences cIdx[])
    scan[t] = cnt;
    __syncthreads();
    for (int o = 1; o < MOLSIZE; o <<= 1) {
        int v = 0; if (t >= o) v = scan[t - o];
        __syncthreads();
        scan[t] += v;
        __syncthreads();
    }
    int p = pairBase[m] + (scan[t] - cnt);

    // ---- emit pair records in (i, then j) order ----
    const int flatI = m * MOLSIZE + t;
    for (int wd = 0; wd < 4; ++wd) {
        unsigned bits = words[wd];
        while (bits) {
            const int j = (wd << 5) + __builtin_ctz(bits);
            bits &= bits - 1;
            const float dx = X[j][0] - xi;
            const float dy = X[j][1] - yi;
            const float dz = X[j][2] - zi;
            const float dist = sqrtf(dx * dx + dy * dy + dz * dz);
            const float inv  = 1.0f / dist;
            out[offMask  + p] = (float)(flatI * MOLSIZE + j);
            out[offMaskL + p] = (float)((m * MOLSIZE + j) * MOLSIZE + t);
            out[offPmol  + p] = (float)m;
            out[offNi    + p] = (float)s;
            out[offNj    + p] = (float)spec[j];
            out[offIdxi  + p] = (float)cIdx[t];
            out[offIdxj  + p] = (float)cIdx[j];
            out[offXij + 3 * p + 0] = dx * inv;
            out[offXij + 3 * p + 1] = dy * inv;
            out[offXij + 3 * p + 2] = dz * inv;
            out[offRij   + p] = dist * LCF_F;
            ++p;
        }
    }
}

// ---------------------------------------------------------------------------
extern "C" void kernel_launch(void* const* d_in, const int* in_sizes, int n_in,
                              void* d_out, int out_size, void* d_ws, size_t ws_size,
                              hipStream_t stream)
{
    (void)in_sizes; (void)n_in; (void)out_size; (void)ws_size;
    const int*   species = (const int*)  d_in[0];
    const float* coords  = (const float*)d_in[1];
    const int*   totq    = (const int*)  d_in[2];
    const float* tore    = (const float*)d_in[3];
    float*       out     = (float*)d_out;

    // workspace carve-up: bitmask (1 MB) + counts/bases/totals
    const size_t BM_WORDS = (size_t)NMOL * MOLSIZE * 4;
    unsigned* bm       = (unsigned*)d_ws;
    int*      pairCnt  = (int*)((char*)d_ws + BM_WORDS * sizeof(unsigned));
    int*      atomCnt  = pairCnt  + NMOL;
    int*      pairBase = atomCnt  + NMOL;
    int*      atomBase = pairBase + NMOL;
    int*      totals   = atomBase + NMOL;

    parser_k1<<<NMOL, MOLSIZE, 0, stream>>>(species, coords, totq, tore,
                                            out, bm, pairCnt, atomCnt);
    parser_k2<<<1, NMOL, 0, stream>>>(out, pairCnt, atomCnt,
                                      pairBase, atomBase, totals);
    parser_k3<<<NMOL, MOLSIZE, 0, stream>>>(species, coords, out, bm,
                                            pairBase, atomBase, totals);
}